// PCAPassConv_81329500717452
// MI455X (gfx1250) — compile-verified
//
#include <hip/hip_runtime.h>
#include <hip/hip_bf16.h>

#define N_NODES 50000
#define N_EDGES 800000
#define D_IN    64
#define D_OUT   64

typedef __attribute__((ext_vector_type(2))) float v2f;
typedef __attribute__((ext_vector_type(8))) float v8f;

// Row stride for paired-W LDS tile: 80 float2 = 160 banks == 32 (mod 64), so
// lanes 0-15 (row kp) and lanes 16-31 (row kp+1) hit disjoint 32-bank halves.
#define WP_STRIDE 80

// ---------------------------------------------------------------------------
// Kernel 1: zero the workspace (neigh_sum + deg)
// ---------------------------------------------------------------------------
__global__ void sage_zero_ws(float* __restrict__ ws, int n) {
    int stride = gridDim.x * blockDim.x;
    for (int i = blockIdx.x * blockDim.x + threadIdx.x; i < n; i += stride)
        ws[i] = 0.0f;
}

// ---------------------------------------------------------------------------
// Kernel 2: edge scatter-add. One wave32 per edge; lane L handles dims L and
// L+32 with global f32 atomics (neigh_sum is L2-resident: 12.8 MB << 192 MB).
// ---------------------------------------------------------------------------
__global__ void sage_edge_scatter(const float* __restrict__ feat,
                                  const long long* __restrict__ src,
                                  const long long* __restrict__ dst,
                                  float* __restrict__ neigh_sum,
                                  float* __restrict__ deg,
                                  int n_edges) {
    int lane   = threadIdx.x & 31;
    int wid    = (blockIdx.x * blockDim.x + threadIdx.x) >> 5;
    int nwaves = (gridDim.x * blockDim.x) >> 5;
    for (int e = wid; e < n_edges; e += nwaves) {
        int s = (int)src[e];
        int d = (int)dst[e];
        const float* fs = feat + (size_t)s * D_IN;
        float*       ns = neigh_sum + (size_t)d * D_IN;
        atomicAdd(ns + lane,      fs[lane]);
        atomicAdd(ns + lane + 32, fs[lane + 32]);
        if (lane == 0) atomicAdd(deg + d, 1.0f);
    }
}

// ---------------------------------------------------------------------------
// Kernel 3: neigh = neigh_sum / max(deg, 1)
// ---------------------------------------------------------------------------
__global__ void sage_normalize(float* __restrict__ neigh,
                               const float* __restrict__ deg, int n) {
    int stride = gridDim.x * blockDim.x;
    for (int i = blockIdx.x * blockDim.x + threadIdx.x; i < n; i += stride) {
        float d = fmaxf(deg[i >> 6], 1.0f);
        neigh[i] = neigh[i] / d;
    }
}

// ---------------------------------------------------------------------------
// Kernel 4: out = relu([feat | neigh] @ W^T) via V_WMMA_F32_16X16X4_F32.
// 256 threads = 8 wave32s; each wave owns 16 node rows and 4 N-tiles (64
// output cols); K = 128 -> 32 WMMA k-steps (16 over feat, 16 over neigh).
// W staged in LDS pre-paired: Wp[kp][n] = {W[n][2kp], W[n][2kp+1]} so each
// B fragment is one aligned ds_load_b64 (kb is always even).
// ---------------------------------------------------------------------------
__device__ __forceinline__ void wmma_khalf(const float* __restrict__ abase,
                                           const v2f (*Wp)[WP_STRIDE],
                                           int kpofs,      // 0 (feat) or 32 (neigh)
                                           int half, int ncol,
                                           v8f acc[4]) {
    for (int ks = 0; ks < 16; ++ks) {
        // A fragment: lane holds A[m][klo], A[m][klo+1] at immediate offset
        v2f a = *(const v2f*)(abase + (ks << 2));
        int kp = kpofs + (ks << 1) + half;   // paired-K row of B fragment
#pragma unroll
        for (int t = 0; t < 4; ++t) {
            v2f b = Wp[kp][(t << 4) + ncol];
            acc[t] = __builtin_amdgcn_wmma_f32_16x16x4_f32(
                false, a, false, b, (short)0, acc[t], false, false);
        }
    }
}

__launch_bounds__(256)
__global__ void sage_gemm_wmma(const float* __restrict__ feat,
                               const float* __restrict__ neigh,
                               const float* __restrict__ W,   // [64][128] row-major
                               float* __restrict__ out) {     // [N][64]
    __shared__ v2f Wp[64][WP_STRIDE];   // Wp[kp][n] = {W[n][2kp], W[n][2kp+1]}

    int tid = threadIdx.x;
    float* wps = (float*)Wp;
    for (int i = tid; i < D_OUT * 2 * D_IN; i += 256) {
        int n = i >> 7;      // output col 0..63
        int k = i & 127;     // input dim 0..127
        wps[(size_t)(k >> 1) * (2 * WP_STRIDE) + (n << 1) + (k & 1)] = W[i];
    }
    __syncthreads();

    int lane = tid & 31;
    int wave = tid >> 5;
    int nodeBase = blockIdx.x * 128 + wave * 16;
    if (nodeBase >= N_NODES) return;   // wave-uniform exit (no syncs below)

    int half = lane >> 4;        // 0: K lo pair, 1: K hi pair (A/B frag layout)
    int ncol = lane & 15;
    int m = nodeBase + ncol;     // A-matrix row for this lane
    if (m >= N_NODES) m = N_NODES - 1;   // clamp tail loads; stores are guarded

    v8f acc[4] = {};   // four 16x16 f32 C/D tiles (cols 0-15,16-31,32-47,48-63)

    const float* afeat  = feat  + (size_t)m * D_IN + (half << 1);
    const float* aneigh = neigh + (size_t)m * D_IN + (half << 1);
    wmma_khalf(afeat,  Wp,  0, half, ncol, acc);   // K = 0..63   (self feats)
    wmma_khalf(aneigh, Wp, 32, half, ncol, acc);   // K = 64..127 (neighbor mean)

    // C/D tile layout: VGPR r -> row r (lanes 0-15) / row r+8 (lanes 16-31)
    float* obase = out + (size_t)(nodeBase + (half << 3)) * D_OUT + ncol;
    if (nodeBase + 16 <= N_NODES) {
        // Fast path: 32 plain stores, offsets fold into IOFFSET
#pragma unroll
        for (int t = 0; t < 4; ++t)
#pragma unroll
            for (int r = 0; r < 8; ++r)
                obase[(size_t)(r * D_OUT) + (t << 4)] = fmaxf(acc[t][r], 0.0f);
    } else {
        // Tail (last wave of last block only): per-row guard
#pragma unroll
        for (int t = 0; t < 4; ++t)
#pragma unroll
            for (int r = 0; r < 8; ++r)
                if (nodeBase + (half << 3) + r < N_NODES)
                    obase[(size_t)(r * D_OUT) + (t << 4)] = fmaxf(acc[t][r], 0.0f);
    }
}

// ---------------------------------------------------------------------------
extern "C" void kernel_launch(void* const* d_in, const int* in_sizes, int n_in,
                              void* d_out, int out_size, void* d_ws, size_t ws_size,
                              hipStream_t stream) {
    const float*     feat = (const float*)d_in[0];
    const long long* ei   = (const long long*)d_in[1];   // [2][N_EDGES], int64
    const float*     W    = (const float*)d_in[2];       // [64][128]
    float*           outp = (float*)d_out;

    const long long* src = ei;
    const long long* dst = ei + N_EDGES;

    float* neigh = (float*)d_ws;                         // [N_NODES][64]
    float* deg   = neigh + (size_t)N_NODES * D_IN;       // [N_NODES]

    int ws_elems = N_NODES * D_IN + N_NODES;
    sage_zero_ws<<<1024, 256, 0, stream>>>(neigh, ws_elems);

    sage_edge_scatter<<<4096, 256, 0, stream>>>(feat, src, dst, neigh, deg, N_EDGES);

    sage_normalize<<<2048, 256, 0, stream>>>(neigh, deg, N_NODES * D_IN);

    int gemm_blocks = (N_NODES + 127) / 128;
    sage_gemm_wmma<<<gemm_blocks, 256, 0, stream>>>(feat, neigh, W, outp);
}